// ProposalLayer_18013092840121
// MI455X (gfx1250) — compile-verified
//
#include <hip/hip_runtime.h>
#include <cstdint>

// ---------------- problem constants ----------------
constexpr int       BATCH = 4;
constexpr long long NN    = 261888;   // anchors per batch
constexpr int       KSEL  = 6000;     // PRE_NMS_LIMIT
constexpr int       NOUT  = 1000;     // PROPOSAL_COUNT
constexpr int       CAND  = 8192;     // candidate buffer (pow2 for bitonic)
constexpr int       NBINS = 2048;     // 11-bit radix bins
constexpr float     NMS_THR = 0.7f;

// workspace layout (u32 element offsets)
constexpr size_t OFF_H1   = 0;                               // hist level1 : B*2048
constexpr size_t OFF_H2   = OFF_H1  + (size_t)BATCH * NBINS; // hist level2 : B*2048
constexpr size_t OFF_META = OFF_H2  + (size_t)BATCH * NBINS; // meta        : B*4 (b1, above1, thresh22, count)
constexpr size_t OFF_CK   = OFF_META + (size_t)BATCH * 4;    // cand keys   : B*CAND
constexpr size_t OFF_CI   = OFF_CK  + (size_t)BATCH * CAND;  // cand idx    : B*CAND
constexpr size_t OFF_SI   = OFF_CI  + (size_t)BATCH * CAND;  // sorted idx  : B*KSEL
constexpr size_t OFF_BX   = OFF_SI  + (size_t)BATCH * KSEL;  // boxes(f32)  : B*KSEL*4
constexpr size_t ZERO_BYTES = (OFF_META + BATCH * 4) * 4;    // zero hists + meta

typedef __attribute__((ext_vector_type(2))) float    v2f;
typedef __attribute__((ext_vector_type(8))) float    v8f;
typedef __attribute__((ext_vector_type(4))) unsigned uv4;
typedef __attribute__((ext_vector_type(8))) unsigned uv8;

// ---------------- 1) level-1 histogram of score bit patterns ----------------
__global__ void k_hist1(const float* __restrict__ probs, unsigned* __restrict__ h1) {
  __shared__ unsigned h[NBINS];
  for (int i = threadIdx.x; i < NBINS; i += blockDim.x) h[i] = 0u;
  __syncthreads();
  const int b = blockIdx.y;
  for (long long i = (long long)blockIdx.x * blockDim.x + threadIdx.x; i < NN;
       i += (long long)gridDim.x * blockDim.x) {
    float s = probs[((size_t)b * NN + i) * 2 + 1];
    unsigned key = __float_as_uint(s);            // positive floats: monotonic
    atomicAdd(&h[key >> 21], 1u);
  }
  __syncthreads();
  for (int i = threadIdx.x; i < NBINS; i += blockDim.x)
    if (h[i]) atomicAdd(&h1[(size_t)b * NBINS + i], h[i]);
}

// ---------------- 2) pick level-1 threshold bin ----------------
__global__ void k_sel1(const unsigned* __restrict__ h1, unsigned* __restrict__ meta) {
  const int b = blockIdx.x;
  if (threadIdx.x == 0) {
    unsigned cum = 0, bin = 0, above = 0;
    for (int i = NBINS - 1; i >= 0; --i) {
      unsigned c = h1[(size_t)b * NBINS + i];
      if (cum + c >= (unsigned)KSEL) { bin = (unsigned)i; above = cum; break; }
      cum += c;
    }
    meta[b * 4 + 0] = bin;
    meta[b * 4 + 1] = above;
  }
}

// ---------------- 3) level-2 histogram inside threshold bin ----------------
__global__ void k_hist2(const float* __restrict__ probs, const unsigned* __restrict__ meta,
                        unsigned* __restrict__ h2) {
  __shared__ unsigned h[NBINS];
  for (int i = threadIdx.x; i < NBINS; i += blockDim.x) h[i] = 0u;
  __syncthreads();
  const int b = blockIdx.y;
  const unsigned b1 = meta[b * 4 + 0];
  for (long long i = (long long)blockIdx.x * blockDim.x + threadIdx.x; i < NN;
       i += (long long)gridDim.x * blockDim.x) {
    unsigned key = __float_as_uint(probs[((size_t)b * NN + i) * 2 + 1]);
    if ((key >> 21) == b1) atomicAdd(&h[(key >> 10) & (NBINS - 1)], 1u);
  }
  __syncthreads();
  for (int i = threadIdx.x; i < NBINS; i += blockDim.x)
    if (h[i]) atomicAdd(&h2[(size_t)b * NBINS + i], h[i]);
}

// ---------------- 4) 22-bit threshold prefix ----------------
__global__ void k_sel2(const unsigned* __restrict__ h2, unsigned* __restrict__ meta) {
  const int b = blockIdx.x;
  if (threadIdx.x == 0) {
    unsigned above1 = meta[b * 4 + 1];
    unsigned rem = (unsigned)KSEL - above1;
    unsigned cum = 0, bin = 0;
    for (int i = NBINS - 1; i >= 0; --i) {
      unsigned c = h2[(size_t)b * NBINS + i];
      if (cum + c >= rem) { bin = (unsigned)i; break; }
      cum += c;
    }
    meta[b * 4 + 2] = (meta[b * 4 + 0] << 11) | bin;  // 22-bit prefix
    meta[b * 4 + 3] = 0u;                              // candidate counter
  }
}

// ---------------- 5) compact candidates >= threshold prefix ----------------
__global__ void k_compact(const float* __restrict__ probs, unsigned* __restrict__ meta,
                          unsigned* __restrict__ ck, unsigned* __restrict__ ci) {
  const int b = blockIdx.y;
  const unsigned thr = meta[b * 4 + 2];
  for (long long i = (long long)blockIdx.x * blockDim.x + threadIdx.x; i < NN;
       i += (long long)gridDim.x * blockDim.x) {
    unsigned key = __float_as_uint(probs[((size_t)b * NN + i) * 2 + 1]);
    if ((key >> 10) >= thr) {
      unsigned p = atomicAdd(&meta[b * 4 + 3], 1u);
      if (p < (unsigned)CAND) {
        ck[(size_t)b * CAND + p] = key;
        ci[(size_t)b * CAND + p] = (unsigned)i;
      }
    }
  }
}

// ---------------- 6) bitonic sort candidates, emit sorted top-KSEL indices ----------------
__global__ __launch_bounds__(1024) void k_sortsel(const unsigned* __restrict__ ck,
                                                  const unsigned* __restrict__ ci,
                                                  const unsigned* __restrict__ meta,
                                                  unsigned* __restrict__ sidx) {
  __shared__ unsigned long long s[CAND];   // 64 KB
  const int b = blockIdx.x;
  unsigned cnt = meta[b * 4 + 3];
  if (cnt > (unsigned)CAND) cnt = CAND;
  for (int t = threadIdx.x; t < CAND; t += blockDim.x) {
    unsigned long long v = 0ull;
    if (t < (int)cnt) {
      unsigned key = ck[(size_t)b * CAND + t];
      unsigned id  = ci[(size_t)b * CAND + t];
      v = ((unsigned long long)key << 32) | (unsigned long long)(~id); // desc key, asc idx
    }
    s[t] = v;
  }
  __syncthreads();
  for (unsigned kk = 2; kk <= (unsigned)CAND; kk <<= 1) {
    for (unsigned j = kk >> 1; j > 0; j >>= 1) {
      for (unsigned i = threadIdx.x; i < (unsigned)CAND; i += blockDim.x) {
        unsigned ixj = i ^ j;
        if (ixj > i) {
          unsigned long long a = s[i], c = s[ixj];
          bool desc = ((i & kk) == 0);
          if (desc ? (a < c) : (a > c)) { s[i] = c; s[ixj] = a; }
        }
      }
      __syncthreads();
    }
  }
  for (int t = threadIdx.x; t < KSEL; t += blockDim.x)
    sidx[(size_t)b * KSEL + t] = ~(unsigned)(s[t] & 0xFFFFFFFFull);
}

// ---------------- 7) gather + box-delta decode + clip ----------------
__global__ void k_decode(const float* __restrict__ deltas, const float* __restrict__ anchors,
                         const unsigned* __restrict__ sidx, float* __restrict__ boxesG) {
  const int b = blockIdx.y;
  const int t = blockIdx.x * blockDim.x + threadIdx.x;
  if (t >= KSEL) return;
  unsigned i = sidx[(size_t)b * KSEL + t];
  size_t o = ((size_t)b * NN + i) * 4;
  float dy = deltas[o + 0] * 0.1f, dx = deltas[o + 1] * 0.1f;
  float dh = deltas[o + 2] * 0.2f, dw = deltas[o + 3] * 0.2f;
  float y1 = anchors[o + 0], x1 = anchors[o + 1];
  float y2 = anchors[o + 2], x2 = anchors[o + 3];
  float h = y2 - y1, w = x2 - x1;
  float cy = y1 + 0.5f * h + dy * h;
  float cx = x1 + 0.5f * w + dx * w;
  h *= expf(dh);  w *= expf(dw);
  float ny1 = cy - 0.5f * h, nx1 = cx - 0.5f * w;
  float ny2 = ny1 + h, nx2 = nx1 + w;
  ny1 = fminf(fmaxf(ny1, 0.f), 1.f);  nx1 = fminf(fmaxf(nx1, 0.f), 1.f);
  ny2 = fminf(fmaxf(ny2, 0.f), 1.f);  nx2 = fminf(fmaxf(nx2, 0.f), 1.f);
  size_t oo = ((size_t)b * KSEL + t) * 4;
  boxesG[oo + 0] = ny1; boxesG[oo + 1] = nx1; boxesG[oo + 2] = ny2; boxesG[oo + 3] = nx2;
}

// ---------------- 8) chunked greedy NMS (WMMA area sums + TDM chunk staging) + output ----------------
__global__ __launch_bounds__(512) void k_nms_out(const float* __restrict__ boxesG,
                                                 float* __restrict__ out) {
  constexpr int NCH = KSEL / 16;   // 375 chunks of 16
  __shared__ float    sPad[64];                // sacrificial pad at low LDS offsets
  __shared__ float    sCB[64];                 // current chunk boxes (16 x 4)
  __shared__ float    sCA[16];                 // current chunk areas
  __shared__ unsigned sSup[(KSEL + 31) / 32];  // suppression bitmask
  __shared__ unsigned sKeep;                   // survivors of current chunk (16-bit mask)
  __shared__ int      sSel[NOUT];
  __shared__ int      sCnt;

  const int b    = blockIdx.x;
  const int tid  = threadIdx.x;
  const int lane = tid & 31;
  const int wave = tid >> 5;
  const int nwaves = (int)blockDim.x >> 5;
  const float4* bg4 = reinterpret_cast<const float4*>(boxesG) + (size_t)b * KSEL;

  for (int i = tid; i < (KSEL + 31) / 32; i += blockDim.x) sSup[i] = 0u;
  if (tid == 0) sPad[0] = 0.f;
  __syncthreads();

  for (int c = 0; c < NCH; ++c) {
    const int base = c * 16;

    // --- stage chunk boxes into LDS via Tensor Data Mover (wave 0 issues) ---
    if (tid < 32) {
      unsigned ldsAddr = (unsigned)(unsigned long long)(uintptr_t)&sCB[0];
      unsigned long long ga =
          (unsigned long long)(uintptr_t)(boxesG + ((size_t)b * KSEL + base) * 4);
      // D# group0: count=1 | lds_addr | global_addr[56:0] | type=2
      uv4 g0 = {1u, ldsAddr, (unsigned)ga,
                ((unsigned)(ga >> 32) & 0x01FFFFFFu) | (2u << 30)};
      // D# group1: data_size=4B; tensor_dim0=64, tensor_dim1=1, tile_dim0=64, stride0=64
      uv8 g1 = {(2u << 16), (64u << 16), (1u << 16), (64u << 16), 0u, 64u, 0u, 0u};
      asm volatile("tensor_load_to_lds %0, %1" :: "s"(g0), "s"(g1) : "memory");
      __builtin_amdgcn_s_wait_tensorcnt(0);
      // prefetch next chunk's boxes into cache while this chunk is processed
      if (c + 1 < NCH)
        __builtin_prefetch(boxesG + ((size_t)b * KSEL + base + 16) * 4, 0, 0);
    }
    __syncthreads();
    // belt-and-suspenders: plain load of the same 256B (idempotent if TDM was correct)
    if (tid < 64) sCB[tid] = boxesG[((size_t)b * KSEL + base) * 4 + tid];
    __syncthreads();
    if (tid < 16)
      sCA[tid] = (sCB[tid * 4 + 2] - sCB[tid * 4 + 0]) *
                 (sCB[tid * 4 + 3] - sCB[tid * 4 + 1]);
    __syncthreads();

    // --- intra-chunk sequential greedy (16 boxes; thread 0, division-free) ---
    if (tid == 0) {
      unsigned wd = sSup[base >> 5];
      unsigned sh = base & 31;               // 0 or 16
      unsigned cur = (wd >> sh) & 0xFFFFu;
      unsigned keep = 0u;
      for (int ii = 0; ii < 16; ++ii) {
        if ((cur >> ii) & 1u) continue;
        keep |= 1u << ii;
        for (int jj = ii + 1; jj < 16; ++jj) {
          if ((cur >> jj) & 1u) continue;
          float ih = fminf(sCB[ii * 4 + 2], sCB[jj * 4 + 2]) -
                     fmaxf(sCB[ii * 4 + 0], sCB[jj * 4 + 0]);
          float iw = fminf(sCB[ii * 4 + 3], sCB[jj * 4 + 3]) -
                     fmaxf(sCB[ii * 4 + 1], sCB[jj * 4 + 1]);
          float inter = fmaxf(ih, 0.f) * fmaxf(iw, 0.f);
          float uni = sCA[ii] + sCA[jj] - inter + 1e-8f;   // > 0 always
          if (inter > NMS_THR * uni) cur |= 1u << jj;      // == (iou > THR)
        }
      }
      sSup[base >> 5] = wd | (cur << sh);
      sKeep = keep;
    }
    __syncthreads();

    // --- forward suppression: chunk survivors vs all later 16-col tiles ---
    const unsigned keepm = sKeep;     // block-uniform
    if (keepm) {
      const int ib = (lane >= 16) ? 8 : 0;
      // chunk-invariant per-lane state (hoisted out of tile loop)
      const float ai = sCA[lane & 15];          // valid LDS read for all lanes
      v2f av;
      av.x = (lane < 16) ? ai : 0.f;            // A: K0 rows in lanes 0-15
      av.y = (lane < 16) ? 1.f : 0.f;           //    K1; lanes 16-31 hold K2,K3 = 0
      const unsigned keephalf = (keepm >> ib) & 0xFFu;
      float4 rb[8];
#pragma unroll
      for (int v = 0; v < 8; ++v) rb[v] = ((const float4*)sCB)[ib + v];

      for (int tb = c + 1 + wave; tb < NCH; tb += nwaves) {
        const int j = tb * 16 + (lane & 15);
        float4 bj = bg4[j];
        float areaj = (bj.z - bj.x) * (bj.w - bj.y);
        // WMMA: D[i][j] = area_i * 1 + 1 * 0 = area_i (area_j meets A's K2 = 0)
        v2f bv;
        bv.x = (lane < 16) ? 1.f : areaj;       // B: row K0 = 1, row K2 = area_j (lanes 16-31)
        bv.y = 0.f;                             //    rows K1,K3 = 0
        v8f cc = {0.f, 0.f, 0.f, 0.f, 0.f, 0.f, 0.f, 0.f};
        cc = __builtin_amdgcn_wmma_f32_16x16x4_f32(false, av, false, bv,
                                                   (short)0, cc, false, false);
        unsigned sup = 0u;                      // branchless, division-free
#pragma unroll
        for (int v = 0; v < 8; ++v) {
          float ih = fminf(rb[v].z, bj.z) - fmaxf(rb[v].x, bj.x);
          float iw = fminf(rb[v].w, bj.w) - fmaxf(rb[v].y, bj.y);
          float inter = fmaxf(ih, 0.f) * fmaxf(iw, 0.f);
          float uni = cc[v] - inter + 1e-8f;    // cc[v] = area_i (area_j term is 0)
          unsigned kb = (keephalf >> v) & 1u;
          sup |= kb & (unsigned)(inter > NMS_THR * uni);
        }
        if (sup) atomicOr(&sSup[(unsigned)j >> 5], 1u << ((unsigned)j & 31u));
      }
    }
    __syncthreads();
  }

  // --- ordered compaction of kept boxes, zero-padded to NOUT ---
  if (tid == 0) {
    int cnt = 0;
    for (int j = 0; j < KSEL && cnt < NOUT; ++j)
      if (!((sSup[j >> 5] >> (j & 31)) & 1u)) sSel[cnt++] = j;
    sCnt = cnt;
  }
  __syncthreads();
  const int cnt = sCnt;
  for (int p = tid; p < NOUT; p += blockDim.x) {
    float4 v = make_float4(0.f, 0.f, 0.f, 0.f);
    if (p < cnt) v = bg4[sSel[p]];
    float* o = out + ((size_t)b * NOUT + p) * 4;
    o[0] = v.x; o[1] = v.y; o[2] = v.z; o[3] = v.w;
  }
}

// ---------------- host side ----------------
extern "C" void kernel_launch(void* const* d_in, const int* in_sizes, int n_in,
                              void* d_out, int out_size, void* d_ws, size_t ws_size,
                              hipStream_t stream) {
  const float* probs   = (const float*)d_in[0];  // (B,N,2)
  const float* deltas  = (const float*)d_in[1];  // (B,N,4)
  const float* anchors = (const float*)d_in[2];  // (B,N,4)
  float*       out     = (float*)d_out;          // (B,1000,4)
  unsigned*    W       = (unsigned*)d_ws;
  float*       Wf      = (float*)d_ws;

  hipMemsetAsync(d_ws, 0, ZERO_BYTES, stream);   // zero hists + meta each call

  dim3 gridN(256, BATCH);
  k_hist1  <<<gridN, 256, 0, stream>>>(probs, W + OFF_H1);
  k_sel1   <<<BATCH, 32, 0, stream>>>(W + OFF_H1, W + OFF_META);
  k_hist2  <<<gridN, 256, 0, stream>>>(probs, W + OFF_META, W + OFF_H2);
  k_sel2   <<<BATCH, 32, 0, stream>>>(W + OFF_H2, W + OFF_META);
  k_compact<<<gridN, 256, 0, stream>>>(probs, W + OFF_META, W + OFF_CK, W + OFF_CI);
  k_sortsel<<<BATCH, 1024, 0, stream>>>(W + OFF_CK, W + OFF_CI, W + OFF_META, W + OFF_SI);
  dim3 gridK((KSEL + 255) / 256, BATCH);
  k_decode <<<gridK, 256, 0, stream>>>(deltas, anchors, W + OFF_SI, Wf + OFF_BX);
  k_nms_out<<<BATCH, 512, 0, stream>>>(Wf + OFF_BX, out);
}